// SelectBlock_8658654068996
// MI455X (gfx1250) — compile-verified
//
#include <hip/hip_runtime.h>
#include <math.h>

// Problem constants (from reference)
#define B_    64
#define T_    60
#define L_    49
#define EMB_  768
#define HID_  768
#define VOC_  10000
#define X2E_  1536   // 2*EMBED

typedef __attribute__((ext_vector_type(2))) float v2f;
typedef __attribute__((ext_vector_type(8))) float v8f;

// ---------------------------------------------------------------------------
// Generic fp32 WMMA GEMM:  C[M,N] = A[M,K] * W[N,K]^T  (+ C)  (+ bias[n])
//   - 128 threads = 4 waves; block tile 64x64; each wave owns a 32x32
//     register tile = 2x2 WMMA 16x16 tiles (4 x v8f accumulators)
//   - LDS-staged A/W panels (64 x 32 floats, padded rows for bank spread)
//   - shiftMod > 0: A row r reads A[r-1], rows with r%shiftMod==0 are zero
//     (implements hiddens_{t-1} without materializing it)
//   - M must be a multiple of 64 and K a multiple of 32 (true for all calls);
//     N edge handled by clamped loads + guarded stores.
// ---------------------------------------------------------------------------
__global__ void __launch_bounds__(128)
wmma_gemm_f32(const float* __restrict__ A, int lda,
              const float* __restrict__ W, int ldw,
              float* __restrict__ C, int ldc,
              int N, int K,
              const float* __restrict__ bias,
              int addC, int shiftMod) {
  __shared__ __align__(16) float sA[64][36];
  __shared__ __align__(16) float sW[64][36];

  const int tid  = threadIdx.x;
  const int lane = tid & 31;
  const int wid  = tid >> 5;   // 0..3
  const int wm   = wid >> 1;   // 0..1 (M 32-sub-tile)
  const int wn   = wid & 1;    // 0..1 (N 32-sub-tile)

  const int m0 = blockIdx.y * 64;
  const int n0 = blockIdx.x * 64;

  // panel loader mapping: each thread fills rows lr+{0,16,32,48}, 4 cols
  const int lr = tid >> 3;        // 0..15
  const int lc = (tid & 7) * 4;   // 0,4,...,28

  v8f acc00 = {}, acc01 = {}, acc10 = {}, acc11 = {};

  for (int kb = 0; kb < K; kb += 32) {
#pragma unroll
    for (int half = 0; half < 4; ++half) {
      const int rr = lr + half * 16;
      // ---- A panel ----
      const int grow = m0 + rr;
      float4 av;
      if (shiftMod && (grow % shiftMod) == 0) {
        av = make_float4(0.f, 0.f, 0.f, 0.f);
      } else {
        const int srow = shiftMod ? (grow - 1) : grow;
        av = *(const float4*)(A + (size_t)srow * lda + kb + lc);
      }
      *(float4*)&sA[rr][lc] = av;
      // ---- W panel (clamp OOB rows; those columns are masked at store) ----
      int wrow = n0 + rr;
      if (wrow >= N) wrow = N - 1;
      *(float4*)&sW[rr][lc] = *(const float4*)(W + (size_t)wrow * ldw + kb + lc);
    }
    __syncthreads();

    // A 16x4 fragment: lanes 0-15 -> K {0,1}, lanes 16-31 -> K {2,3}
    const int fr   = lane & 15;
    const int koff = (lane >> 4) * 2;
    const float* aBase0 = &sA[wm * 32 + fr][koff];
    const float* aBase1 = &sA[wm * 32 + 16 + fr][koff];
    const float* bBase0 = &sW[wn * 32 + fr][koff];
    const float* bBase1 = &sW[wn * 32 + 16 + fr][koff];
#pragma unroll
    for (int kk = 0; kk < 32; kk += 4) {
      v2f a0 = *(const v2f*)(aBase0 + kk);
      v2f a1 = *(const v2f*)(aBase1 + kk);
      v2f b0 = *(const v2f*)(bBase0 + kk);
      v2f b1 = *(const v2f*)(bBase1 + kk);
      acc00 = __builtin_amdgcn_wmma_f32_16x16x4_f32(false, a0, false, b0, (short)0, acc00, false, false);
      acc01 = __builtin_amdgcn_wmma_f32_16x16x4_f32(false, a0, false, b1, (short)0, acc01, false, false);
      acc10 = __builtin_amdgcn_wmma_f32_16x16x4_f32(false, a1, false, b0, (short)0, acc10, false, false);
      acc11 = __builtin_amdgcn_wmma_f32_16x16x4_f32(false, a1, false, b1, (short)0, acc11, false, false);
    }
    __syncthreads();
  }

  // C/D layout: VGPR r -> row (r + 8*(lane>=16)), col = lane&15
  const int colBase = n0 + wn * 32 + (lane & 15);
  const int rowBase = m0 + wm * 32 + ((lane >> 4) << 3);
  v8f* accs[4] = { &acc00, &acc01, &acc10, &acc11 };
#pragma unroll
  for (int im = 0; im < 2; ++im) {
#pragma unroll
    for (int in = 0; in < 2; ++in) {
      const int nn = colBase + in * 16;
      if (nn < N) {
        const float badd = bias ? bias[nn] : 0.f;
        const v8f& a = *accs[im * 2 + in];
#pragma unroll
        for (int r = 0; r < 8; ++r) {
          const int mm = rowBase + im * 16 + r;
          float v = a[r] + badd;
          if (addC) v += C[(size_t)mm * ldc + nn];
          C[(size_t)mm * ldc + nn] = v;
        }
      }
    }
  }
}

// ---------------------------------------------------------------------------
// xmean[b,e] = mean_t x[b,t,e]   (fold the mean BEFORE the q GEMM)
// ---------------------------------------------------------------------------
__global__ void mean_x_kernel(const float* __restrict__ x, float* __restrict__ xm) {
  const int i = blockIdx.x * blockDim.x + threadIdx.x;
  if (i >= B_ * X2E_) return;
  const int b = i / X2E_, e = i % X2E_;
  const float* p = x + (size_t)b * T_ * X2E_ + e;
  float s = 0.f;
  for (int t = 0; t < T_; ++t) s += p[(size_t)t * X2E_];
  xm[i] = s * (1.0f / T_);
}

// s_t = sigmoid(pre) * tanh(cells)
__global__ void st_kernel(const float* __restrict__ pre,
                          const float* __restrict__ cells,
                          float* __restrict__ st, int n) {
  const int i = blockIdx.x * blockDim.x + threadIdx.x;
  if (i >= n) return;
  const float g = 1.0f / (1.0f + expf(-pre[i]));
  st[i] = g * tanhf(cells[i]);
}

// ---------------------------------------------------------------------------
// Switch: per (b,l): sH = sum_h tanh(aH+q)*wsw ; sG likewise; 2-way softmax;
//         V = w0*H + w1*G  (V may alias aG: each block reads its aG row fully
//         before writing its V row)
// ---------------------------------------------------------------------------
__global__ void __launch_bounds__(256)
switch_kernel(const float* __restrict__ aH, const float* __restrict__ aG,
              const float* __restrict__ q, const float* __restrict__ wsw,
              const float* __restrict__ H, const float* __restrict__ G,
              float* __restrict__ V, float* __restrict__ out_sw) {
  const int bl = blockIdx.x;          // b*L + l
  const int b  = bl / L_;
  __shared__ float redH[256], redG[256];
  __shared__ float w01[2];
  const size_t row = (size_t)bl * HID_;
  const size_t qo  = (size_t)b * HID_;

  float sh = 0.f, sg = 0.f;
  for (int h = threadIdx.x; h < HID_; h += 256) {
    const float qq = q[qo + h], w = wsw[h];
    sh += tanhf(aH[row + h] + qq) * w;
    sg += tanhf(aG[row + h] + qq) * w;
  }
  redH[threadIdx.x] = sh;
  redG[threadIdx.x] = sg;
  __syncthreads();
  for (int s = 128; s > 0; s >>= 1) {
    if (threadIdx.x < s) {
      redH[threadIdx.x] += redH[threadIdx.x + s];
      redG[threadIdx.x] += redG[threadIdx.x + s];
    }
    __syncthreads();
  }
  if (threadIdx.x == 0) {
    const float zh = redH[0], zg = redG[0];
    const float m  = fmaxf(zh, zg);
    const float eh = expf(zh - m), eg = expf(zg - m);
    const float inv = 1.0f / (eh + eg);
    const float w0 = eh * inv, w1 = eg * inv;
    out_sw[(size_t)bl * 2 + 0] = w0;
    out_sw[(size_t)bl * 2 + 1] = w1;
    w01[0] = w0; w01[1] = w1;
  }
  __syncthreads();
  const float w0 = w01[0], w1 = w01[1];
  for (int h = threadIdx.x; h < HID_; h += 256)
    V[row + h] = w0 * H[row + h] + w1 * G[row + h];
}

// ---------------------------------------------------------------------------
// Per (b,t): z_t[l] (49 tanh dot products), z_s; softmax(49) -> alpha;
// softmax(50) -> beta; chh = beta*s_t + (1-beta)*(alpha . V) + hiddens
// pv/pg/ps stored with padded leading dim 64 (only k<49 read).
// ---------------------------------------------------------------------------
__global__ void __launch_bounds__(128)
zt_alpha_kernel(const float* __restrict__ pv, const float* __restrict__ pg,
                const float* __restrict__ ps, const float* __restrict__ wh,
                const float* __restrict__ V, const float* __restrict__ st,
                const float* __restrict__ hiddens,
                float* __restrict__ out_alpha, float* __restrict__ out_beta,
                float* __restrict__ chh) {
  const int bt = blockIdx.x;           // b*T + t
  const int b  = bt / T_;
  __shared__ float z[64];
  __shared__ float salpha[64];
  __shared__ float sbeta;

  const float* pgrow = pg + (size_t)bt * 64;
  const int l = threadIdx.x;
  if (l < L_) {
    const float* pvrow = pv + ((size_t)b * L_ + l) * 64;
    float s = 0.f;
    for (int k = 0; k < L_; ++k) s += tanhf(pvrow[k] + pgrow[k]) * wh[k];
    z[l] = s;
  } else if (l == L_) {
    const float* psrow = ps + (size_t)bt * 64;
    float s = 0.f;
    for (int k = 0; k < L_; ++k) s += tanhf(psrow[k] + pgrow[k]) * wh[k];
    z[L_] = s;
  }
  __syncthreads();

  if (threadIdx.x == 0) {
    float m49 = -1e30f;
    for (int i = 0; i < L_; ++i) m49 = fmaxf(m49, z[i]);
    float s49 = 0.f;
    for (int i = 0; i < L_; ++i) { const float e = expf(z[i] - m49); salpha[i] = e; s49 += e; }
    const float inv49 = 1.0f / s49;
    for (int i = 0; i < L_; ++i) salpha[i] *= inv49;
    const float m50 = fmaxf(m49, z[L_]);
    float s50 = 0.f;
    for (int i = 0; i <= L_; ++i) s50 += expf(z[i] - m50);
    sbeta = expf(z[L_] - m50) / s50;
  }
  __syncthreads();

  if (threadIdx.x < L_) out_alpha[(size_t)bt * L_ + threadIdx.x] = salpha[threadIdx.x];
  if (threadIdx.x == 0) out_beta[bt] = sbeta;

  const float beta = sbeta;
  for (int h = threadIdx.x; h < HID_; h += 128) {
    float ct = 0.f;
    const float* vp = V + (size_t)b * L_ * HID_ + h;
    for (int ll = 0; ll < L_; ++ll) ct += salpha[ll] * vp[(size_t)ll * HID_];
    const size_t idx = (size_t)bt * HID_ + h;
    const float chat = beta * st[idx] + (1.0f - beta) * ct;
    chh[idx] = chat + hiddens[idx];
  }
}

// ---------------------------------------------------------------------------
extern "C" void kernel_launch(void* const* d_in, const int* in_sizes, int n_in,
                              void* d_out, int out_size, void* d_ws, size_t ws_size,
                              hipStream_t stream) {
  (void)in_sizes; (void)n_in; (void)out_size; (void)ws_size;

  const float* x       = (const float*)d_in[0];
  const float* hiddens = (const float*)d_in[1];
  const float* cells   = (const float*)d_in[2];
  const float* G       = (const float*)d_in[3];
  const float* H       = (const float*)d_in[4];
  const float* W_sx    = (const float*)d_in[5];
  const float* W_sh    = (const float*)d_in[6];
  const float* W_ax    = (const float*)d_in[7];
  const float* W_ah    = (const float*)d_in[8];
  const float* W_ag    = (const float*)d_in[9];
  const float* w_sw    = (const float*)d_in[10];
  const float* Wv      = (const float*)d_in[11];
  const float* Wg      = (const float*)d_in[12];
  const float* Ws      = (const float*)d_in[13];
  const float* wh      = (const float*)d_in[14];
  const float* W_mlp   = (const float*)d_in[15];
  const float* b_mlp   = (const float*)d_in[16];

  const int BT = B_ * T_;   // 3840
  const int BL = B_ * L_;   // 3136

  // workspace layout (floats); pre is reused for (c_hat + hiddens)
  float* ws    = (float*)d_ws;
  float* f_st  = ws;                                 // BT*HID
  float* f_pre = f_st  + (size_t)BT * HID_;          // BT*HID
  float* f_xm  = f_pre + (size_t)BT * HID_;          // B*2E
  float* f_q   = f_xm  + (size_t)B_ * X2E_;          // B*HID
  float* f_aH  = f_q   + (size_t)B_ * HID_;          // BL*HID
  float* f_aG  = f_aH  + (size_t)BL * HID_;          // BL*HID (later V)
  float* f_pv  = f_aG  + (size_t)BL * HID_;          // BL*64 (ld 64, 49 valid)
  float* f_pg  = f_pv  + (size_t)BL * 64;            // BT*64
  float* f_ps  = f_pg  + (size_t)BT * 64;            // BT*64

  // outputs, concatenated: scores, alpha, beta, switch_weights
  float* o_scores = (float*)d_out;
  float* o_alpha  = o_scores + (size_t)BT * VOC_;
  float* o_beta   = o_alpha  + (size_t)BT * L_;
  float* o_sw     = o_beta   + (size_t)BT;

  // 1. xmean = mean_t(x)
  mean_x_kernel<<<(B_ * X2E_ + 255) / 256, 256, 0, stream>>>(x, f_xm);
  // 2. q = xmean @ W_ax^T            (64 x 768, K=1536)
  wmma_gemm_f32<<<dim3(HID_ / 64, B_ / 64), 128, 0, stream>>>(
      f_xm, X2E_, W_ax, X2E_, f_q, HID_, HID_, X2E_, nullptr, 0, 0);
  // 3. pre = x @ W_sx^T              (3840 x 768, K=1536)
  wmma_gemm_f32<<<dim3(HID_ / 64, BT / 64), 128, 0, stream>>>(
      x, X2E_, W_sx, X2E_, f_pre, HID_, HID_, X2E_, nullptr, 0, 0);
  // 4. pre += hiddens_{t-1} @ W_sh^T (shifted-A path, addC)
  wmma_gemm_f32<<<dim3(HID_ / 64, BT / 64), 128, 0, stream>>>(
      hiddens, HID_, W_sh, HID_, f_pre, HID_, HID_, HID_, nullptr, 1, T_);
  // 5. s_t = sigmoid(pre) * tanh(cells)
  st_kernel<<<(BT * HID_ + 255) / 256, 256, 0, stream>>>(f_pre, cells, f_st, BT * HID_);
  // 6. aH = H @ W_ah^T               (3136 x 768, K=768)
  wmma_gemm_f32<<<dim3(HID_ / 64, BL / 64), 128, 0, stream>>>(
      H, HID_, W_ah, HID_, f_aH, HID_, HID_, HID_, nullptr, 0, 0);
  // 7. aG = G @ W_ag^T
  wmma_gemm_f32<<<dim3(HID_ / 64, BL / 64), 128, 0, stream>>>(
      G, HID_, W_ag, HID_, f_aG, HID_, HID_, HID_, nullptr, 0, 0);
  // 8. switch softmax + V (V written over aG buffer)
  switch_kernel<<<BL, 256, 0, stream>>>(f_aH, f_aG, f_q, w_sw, H, G, f_aG, o_sw);
  // 9. pv = V @ Wv^T                 (3136 x 49, ld 64)
  wmma_gemm_f32<<<dim3(1, BL / 64), 128, 0, stream>>>(
      f_aG, HID_, Wv, HID_, f_pv, 64, L_, HID_, nullptr, 0, 0);
  // 10. pg = hiddens @ Wg^T          (3840 x 49, ld 64)
  wmma_gemm_f32<<<dim3(1, BT / 64), 128, 0, stream>>>(
      hiddens, HID_, Wg, HID_, f_pg, 64, L_, HID_, nullptr, 0, 0);
  // 11. ps = s_t @ Ws^T              (3840 x 49, ld 64)
  wmma_gemm_f32<<<dim3(1, BT / 64), 128, 0, stream>>>(
      f_st, HID_, Ws, HID_, f_ps, 64, L_, HID_, nullptr, 0, 0);
  // 12. z_t / alpha / beta / chh (chh into f_pre)
  zt_alpha_kernel<<<BT, 128, 0, stream>>>(
      f_pv, f_pg, f_ps, wh, f_aG, f_st, hiddens, o_alpha, o_beta, f_pre);
  // 13. scores = chh @ W_mlp^T + b   (3840 x 10000, K=768)
  wmma_gemm_f32<<<dim3((VOC_ + 63) / 64, BT / 64), 128, 0, stream>>>(
      f_pre, HID_, W_mlp, HID_, o_scores, VOC_, VOC_, HID_, b_mlp, 0, 0);
}